// model_1_TopKPooling_Mod_44813688767087
// MI455X (gfx1250) — compile-verified
//
#include <hip/hip_runtime.h>
#include <math.h>

typedef float v2f __attribute__((ext_vector_type(2)));
typedef float v8f __attribute__((ext_vector_type(8)));

// Problem constants (fixed by the reference setup)
#define NN 131072      // total nodes
#define CC 512         // channels
#define BB 256         // graphs
#define MM 512         // nodes per graph
#define KK 256         // kept nodes per graph
#define EE 4194304     // edges
#define NKEEP (BB * KK)            // 65536

// d_out layout (floats), concatenated in reference return order
#define OFF_X     0LL
#define OFF_EI    ((long long)NKEEP * CC)            // 33554432
#define OFF_BATCH (OFF_EI + 2LL * EE)                // 41943040
#define OFF_PERM  (OFF_BATCH + NKEEP)                // 42008576
#define OFF_MASK  (OFF_PERM + NKEEP)                 // 42074112
#define OFF_SCR   (OFF_MASK + (long long)EE)         // 46268416

// d_ws layout (bytes)
#define WS_SCORE   0                                  // float[NN]
#define WS_PERMI   (NN * 4)                           // int[NKEEP]
#define WS_SCOREP  (WS_PERMI + NKEEP * 4)             // float[NKEEP]
#define WS_MAP     (WS_SCOREP + NKEEP * 4)            // int[NN]
#define WS_INVN    (WS_MAP + NN * 4)                  // float[1]

// ---------------------------------------------------------------------------
// K0: inverse weight norm
// ---------------------------------------------------------------------------
__global__ void k_wnorm(const float* __restrict__ w, float* __restrict__ inv_norm) {
    __shared__ float red[256];
    int t = threadIdx.x;
    float a = w[t];
    float b = w[t + 256];
    red[t] = a * a + b * b;
    __syncthreads();
    for (int s = 128; s > 0; s >>= 1) {
        if (t < s) red[t] += red[t + s];
        __syncthreads();
    }
    if (t == 0) *inv_norm = 1.0f / sqrtf(red[0]);
}

// ---------------------------------------------------------------------------
// K1: score_i = tanh(<x_i, w> * inv_norm) via V_WMMA_F32_16X16X4_F32
// Block: 256 threads = 8 waves, 16 rows per wave -> 128 rows/block.
// A tile (16x4 f32): lane L<16 holds A[L, k..k+1]; lane L>=16 holds A[L-16, k+2..k+3]
// B tile (4x16 f32): replicate w-chunk across all N columns (same lane pattern).
// ---------------------------------------------------------------------------
__global__ void k_score(const float* __restrict__ x, const float* __restrict__ w,
                        const float* __restrict__ inv_norm, float* __restrict__ score) {
    __shared__ float lw[CC];
    int t = threadIdx.x;
    lw[t] = w[t];
    lw[t + 256] = w[t + 256];
    __syncthreads();

    int lane = t & 31;
    int wave = t >> 5;
    int m = lane & 15;       // row within 16-row tile
    int hi = lane >> 4;      // 0: K pair {0,1}, 1: K pair {2,3}
    int row0 = blockIdx.x * 128 + wave * 16;

    const float* xrow = x + (size_t)(row0 + m) * CC + 2 * hi;
    const float* wrow = lw + 2 * hi;

    v8f acc = {};
#pragma unroll 4
    for (int k0 = 0; k0 < CC; k0 += 4) {
        v2f a = *(const v2f*)(xrow + k0);
        v2f b = *(const v2f*)(wrow + k0);
        acc = __builtin_amdgcn_wmma_f32_16x16x4_f32(
            /*neg_a=*/false, a, /*neg_b=*/false, b,
            /*c_mod=*/(short)0, acc, /*reuse_a=*/false, /*reuse_b=*/false);
    }

    // D layout: lanes 0-15 hold rows 0..7 (vgpr r -> row r), lanes 16-31 rows 8..15
    if (m == 0) {
        float inv = *inv_norm;
#pragma unroll
        for (int r = 0; r < 8; ++r) {
            score[row0 + hi * 8 + r] = tanhf(acc[r] * inv);
        }
    }
}

// ---------------------------------------------------------------------------
// K2: per-graph top-K via bitonic sort of 512 u64 keys in LDS.
// key = (~ord(score) << 32) | local_idx  -> ascending sort == descending score,
// stable tie-break on ascending index (matches stable argsort of -score).
// Also fills mapping[] for this graph's 512 exclusively-owned nodes.
// ---------------------------------------------------------------------------
__device__ __forceinline__ unsigned f2ord(float f) {
    unsigned u = __float_as_uint(f);
    return (u & 0x80000000u) ? ~u : (u | 0x80000000u);  // monotonic float->uint
}

__global__ void k_topk(const float* __restrict__ score, const int* __restrict__ batch,
                       float* __restrict__ out_batch, float* __restrict__ out_perm,
                       float* __restrict__ out_scperm,
                       int* __restrict__ perm_i, float* __restrict__ score_perm,
                       int* __restrict__ mapping) {
    __shared__ unsigned long long key[MM];
    int g = blockIdx.x;
    int t = threadIdx.x;  // 256 threads

    for (int i = t; i < MM; i += 256) {
        float s = score[g * MM + i];
        unsigned khi = ~f2ord(s);  // descending score
        key[i] = ((unsigned long long)khi << 32) | (unsigned)i;
    }
    __syncthreads();

    // bitonic sort, ascending, 512 elements, 256 compare-exchange threads
    for (int size = 2; size <= MM; size <<= 1) {
        for (int stride = size >> 1; stride > 0; stride >>= 1) {
            int lo = ((t & ~(stride - 1)) << 1) | (t & (stride - 1));
            int hi = lo + stride;
            bool asc = ((lo & size) == 0);
            unsigned long long a = key[lo], b = key[hi];
            if ((a > b) == asc) { key[lo] = b; key[hi] = a; }
            __syncthreads();
        }
    }

    // kept: sorted positions [0, K)
    {
        int idx_local = (int)(unsigned)(key[t] & 0xFFFFFFFFull);
        int gidx = g * MM + idx_local;
        int kp = g * KK + t;
        float s = score[gidx];
        perm_i[kp] = gidx;
        score_perm[kp] = s;
        mapping[gidx] = kp;
        out_perm[kp] = (float)gidx;
        out_batch[kp] = (float)batch[gidx];
        out_scperm[kp] = s;
    }
    // dropped: sorted positions [K, M)
    {
        int idx_local = (int)(unsigned)(key[t + KK] & 0xFFFFFFFFull);
        mapping[g * MM + idx_local] = -1;
    }
}

// ---------------------------------------------------------------------------
// K3: x_out[r, :] = x[perm[r], :] * score_perm[r]   (float4, 2 rows per block)
// ---------------------------------------------------------------------------
__global__ void k_gather(const float* __restrict__ x, const int* __restrict__ perm_i,
                         const float* __restrict__ score_perm, float* __restrict__ x_out) {
    int row = blockIdx.x * 2 + (threadIdx.x >> 7);
    int j = threadIdx.x & 127;  // 128 float4 per 512-float row
    int p = perm_i[row];
    float s = score_perm[row];
    const float4* xs = (const float4*)x;
    float4 v = xs[(size_t)p * (CC / 4) + j];
    v.x *= s; v.y *= s; v.z *= s; v.w *= s;
    ((float4*)x_out)[(size_t)row * (CC / 4) + j] = v;
}

// ---------------------------------------------------------------------------
// K4: edge remap + mask
// ---------------------------------------------------------------------------
__global__ void k_edges(const int* __restrict__ edge_index, const int* __restrict__ mapping,
                        float* __restrict__ out_row, float* __restrict__ out_col,
                        float* __restrict__ out_mask) {
    int i = blockIdx.x * blockDim.x + threadIdx.x;
    if (i >= EE) return;
    int r = mapping[edge_index[i]];
    int c = mapping[edge_index[EE + i]];
    bool keep = (r >= 0) & (c >= 0);
    out_row[i] = keep ? (float)r : -1.0f;
    out_col[i] = keep ? (float)c : -1.0f;
    out_mask[i] = keep ? 1.0f : 0.0f;
}

// ---------------------------------------------------------------------------
extern "C" void kernel_launch(void* const* d_in, const int* in_sizes, int n_in,
                              void* d_out, int out_size, void* d_ws, size_t ws_size,
                              hipStream_t stream) {
    const float* x          = (const float*)d_in[0];
    const int*   edge_index = (const int*)d_in[1];
    // d_in[2] perm placeholder, d_in[3] j, d_in[4] per_t: j % per_t == 0 -> always recompute
    const int*   batch      = (const int*)d_in[5];
    const float* weight     = (const float*)d_in[6];

    float* out = (float*)d_out;
    char*  ws  = (char*)d_ws;

    float* ws_score  = (float*)(ws + WS_SCORE);
    int*   ws_permi  = (int*)(ws + WS_PERMI);
    float* ws_scperm = (float*)(ws + WS_SCOREP);
    int*   ws_map    = (int*)(ws + WS_MAP);
    float* ws_invn   = (float*)(ws + WS_INVN);

    k_wnorm<<<1, 256, 0, stream>>>(weight, ws_invn);

    k_score<<<NN / 128, 256, 0, stream>>>(x, weight, ws_invn, ws_score);

    k_topk<<<BB, 256, 0, stream>>>(ws_score, batch,
                                   out + OFF_BATCH, out + OFF_PERM, out + OFF_SCR,
                                   ws_permi, ws_scperm, ws_map);

    k_gather<<<NKEEP / 2, 256, 0, stream>>>(x, ws_permi, ws_scperm, out + OFF_X);

    k_edges<<<EE / 256, 256, 0, stream>>>(edge_index, ws_map,
                                          out + OFF_EI, out + OFF_EI + EE, out + OFF_MASK);
}